// PairwiseRankingLoss_53944789238504
// MI455X (gfx1250) — compile-verified
//
#include <hip/hip_runtime.h>

typedef __attribute__((ext_vector_type(2))) float v2f;
typedef __attribute__((ext_vector_type(2))) int v2i;
typedef __attribute__((ext_vector_type(8))) float v8f;

#define MARGIN_F 0.5f
#define EPS_F 1e-6f
#define B_N 8192
#define JT 2048            // j-tile elements staged in LDS (24.6 KB)
#define ROWS_PER_BLOCK 128 // 8 waves * 16 rows
#define THREADS 256
#define NBLOCKS (B_N / ROWS_PER_BLOCK) // 64

// Stage 1: each workgroup owns 128 rows, sweeps all 8192 columns from LDS.
// Masked hinge values are accumulated with V_WMMA_F32_16X16X4_F32 (B = ones),
// i.e. the matrix unit performs the row-wise reduction in full f32.
//
// Note: the reference's off-diagonal mask (i != j) is implied by the rank
// condition targets[i] > targets[i] + EPS being false for all finite inputs,
// so it is not re-checked in the hot loop.
__global__ __launch_bounds__(THREADS) void prl_partial_kernel(
    const float* __restrict__ preds, const float* __restrict__ targets,
    const int* __restrict__ pdb, float* __restrict__ psum,
    float* __restrict__ pcnt) {
  __shared__ float sp[JT]; // preds[j]
  __shared__ float st[JT]; // targets[j] + EPS (pre-biased at stage time)
  __shared__ int sid[JT];  // pdb_ids[j]
  __shared__ float wg_sum[THREADS / 32];
  __shared__ float wg_cnt[THREADS / 32];

  const int tid = threadIdx.x;
  const int wave = tid >> 5;
  const int lane = tid & 31;
  // 16x4 f32 A-matrix layout (wave32): lanes 0-15 -> M=lane, K=0,1 in the two
  // VGPRs; lanes 16-31 -> M=lane-16, K=2,3.
  const int row = lane & 15;
  const int kbase = (lane >> 4) << 1; // 0 or 2
  const int i = blockIdx.x * ROWS_PER_BLOCK + wave * 16 + row;

  const float mpi = MARGIN_F - preds[i]; // hinge = max(mpi + pj, 0)
  const float ti = targets[i];
  const int idi = pdb[i];

  v8f accS = {}; // masked hinge sums (per-row, replicated across N)
  v8f accC = {}; // masked pair counts
  const v2f ones = {1.0f, 1.0f};

  for (int jt = 0; jt < B_N; jt += JT) {
    // Stage the j-tile into LDS (EPS folded into the targets once here,
    // instead of per-pair in the O(B^2) loop).
    for (int k = tid; k < JT; k += THREADS) {
      sp[k] = preds[jt + k];
      st[k] = targets[jt + k] + EPS_F;
      sid[k] = pdb[jt + k];
    }
    __syncthreads();

#pragma unroll 4
    for (int jj = 0; jj < JT; jj += 4) {
      const int l0 = jj + kbase; // even -> 8B-aligned pair loads
      const v2f tj = *(const v2f*)&st[l0];
      const v2f pj = *(const v2f*)&sp[l0];
      const v2i idj = *(const v2i*)&sid[l0];

      // Branchless mask: same pdb && targets[i] > targets[j] + eps.
      const int m0 = (idi == idj.x) & (ti > tj.x);
      const int m1 = (idi == idj.y) & (ti > tj.y);
      const float h0 = fmaxf(mpi + pj.x, 0.0f);
      const float h1 = fmaxf(mpi + pj.y, 0.0f);

      v2f av, cv;
      av.x = m0 ? h0 : 0.0f;
      av.y = m1 ? h1 : 0.0f;
      cv.x = m0 ? 1.0f : 0.0f;
      cv.y = m1 ? 1.0f : 0.0f;

      // D[m][n] += sum_k A[m][k] * 1.0 : matrix-core reduction, f32 exact.
      accS = __builtin_amdgcn_wmma_f32_16x16x4_f32(
          false, av, false, ones, (short)0, accS, false, false);
      accC = __builtin_amdgcn_wmma_f32_16x16x4_f32(
          false, cv, false, ones, (short)0, accC, false, false);
    }
    __syncthreads();
  }

  // 16x16 f32 D layout: lane 0 holds column N=0 for M=0..7 (VGPRs 0..7),
  // lane 16 holds column N=0 for M=8..15. Sum those two lanes only.
  float s = 0.0f, c = 0.0f;
#pragma unroll
  for (int v = 0; v < 8; ++v) {
    s += accS[v];
    c += accC[v];
  }
  const float s_tot = __shfl(s, 0, 32) + __shfl(s, 16, 32);
  const float c_tot = __shfl(c, 0, 32) + __shfl(c, 16, 32);
  if (lane == 0) {
    wg_sum[wave] = s_tot;
    wg_cnt[wave] = c_tot;
  }
  __syncthreads();
  if (tid == 0) {
    float S = 0.0f, C = 0.0f;
#pragma unroll
    for (int w = 0; w < THREADS / 32; ++w) {
      S += wg_sum[w];
      C += wg_cnt[w];
    }
    psum[blockIdx.x] = S;
    pcnt[blockIdx.x] = C;
  }
}

// Stage 2: deterministic fixed-order fold of the 64 block partials.
__global__ void prl_final_kernel(const float* __restrict__ psum,
                                 const float* __restrict__ pcnt,
                                 float* __restrict__ out, int nblocks) {
  if (threadIdx.x == 0 && blockIdx.x == 0) {
    float S = 0.0f, C = 0.0f;
    for (int b = 0; b < nblocks; ++b) {
      S += psum[b];
      C += pcnt[b];
    }
    out[0] = (C == 0.0f) ? 0.0f : (S / fmaxf(C, 1.0f));
  }
}

extern "C" void kernel_launch(void* const* d_in, const int* in_sizes, int n_in,
                              void* d_out, int out_size, void* d_ws,
                              size_t ws_size, hipStream_t stream) {
  (void)in_sizes; (void)n_in; (void)out_size; (void)ws_size;
  const float* preds = (const float*)d_in[0];
  const float* targets = (const float*)d_in[1];
  const int* pdb = (const int*)d_in[2];
  float* out = (float*)d_out;
  float* psum = (float*)d_ws;
  float* pcnt = psum + NBLOCKS;

  prl_partial_kernel<<<dim3(NBLOCKS), dim3(THREADS), 0, stream>>>(
      preds, targets, pdb, psum, pcnt);
  prl_final_kernel<<<dim3(1), dim3(32), 0, stream>>>(psum, pcnt, out, NBLOCKS);
}